// DifferentiableEditLayer_24086176596384
// MI455X (gfx1250) — compile-verified
//
#include <hip/hip_runtime.h>
#include <math.h>

#define DEV __device__ __forceinline__

constexpr int BATCH = 8;
constexpr int H = 1024;
constexpr int W = 1536;
constexpr int P = H * W;             // pixels per batch = 1,572,864
constexpr int TCL = 256;             // tone curve control points
constexpr int RES = 1024;            // tone curve resolution
constexpr int BLOCK = 256;           // 8 waves (wave32)
constexpr int QUADS = P / 4;         // float4 per thread
constexpr int NBLK = QUADS / BLOCK;  // 1536 blocks per batch (exact)

typedef float v4f __attribute__((ext_vector_type(4)));

// ---- feature probes ----------------------------------------------------
#if defined(__has_builtin)
# if __has_builtin(__builtin_amdgcn_rcpf)
#  define HAVE_RCPF 1
# endif
# if __has_builtin(__builtin_amdgcn_global_load_async_to_lds_b128) && \
     __has_builtin(__builtin_amdgcn_s_wait_asynccnt)
#  define USE_ASYNC_LDS 1
# endif
#endif

#ifdef USE_ASYNC_LDS
typedef int v4i __attribute__((ext_vector_type(4)));
typedef v4i __attribute__((address_space(1)))* async_g_t;  // global src
typedef v4i __attribute__((address_space(3)))* async_l_t;  // LDS dst
#endif

// Non-temporal (TH=NT) streaming access: keep the big image streams from
// evicting the L2-resident luma plane (50 MB << 192 MB L2).
DEV v4f ntload4(const float* p) { return __builtin_nontemporal_load((const v4f*)p); }
DEV void ntstore4(float* p, v4f v) { __builtin_nontemporal_store(v, (v4f*)p); }

DEV float clip01(float x) { return fminf(fmaxf(x, 0.0f), 1.0f); }
DEV float clip04(float x) { return fminf(fmaxf(x, 0.0f), 4.0f); }

DEV float frcp(float x) {
#ifdef HAVE_RCPF
  return __builtin_amdgcn_rcpf(x);
#else
  return 1.0f / x;
#endif
}

DEV float fsig(float x) { return frcp(1.0f + __expf(-x)); }

DEV float lumf(float r, float g, float b) {
  return 0.2126f * r + 0.7152f * g + 0.0722f * b;
}

// base pipeline for one channel: clip01 -> WB gain (clip 0..4) -> exposure
// (clip 0..4) -> contrast (clip 0..1)
DEV float basech(float c, float gain, float ex, float cf) {
  c = clip01(c);
  c = clip04(c * gain);
  c = clip04(c * ex);
  c = clip01((c - 0.5f) * cf + 0.5f);
  return c;
}

// fixed-order block reduction (deterministic): wave32 shuffle tree + LDS
DEV void block_reduce_store(float v, float* dst) {
  #pragma unroll
  for (int off = 16; off > 0; off >>= 1) v += __shfl_down(v, off, 32);
  __shared__ float sw[BLOCK / 32];
  const int lane = threadIdx.x & 31, wid = threadIdx.x >> 5;
  if (lane == 0) sw[wid] = v;
  __syncthreads();
  if (threadIdx.x == 0) {
    float t = 0.0f;
    #pragma unroll
    for (int i = 0; i < BLOCK / 32; ++i) t += sw[i];
    *dst = t;
  }
}

// region constants
__constant__ float c_piv[4] = {0.7f, 0.3f, 0.9f, 0.1f};
__constant__ float c_iw[4]  = {1.0f / 0.1f, 1.0f / 0.12f, 1.0f / 0.08f, 1.0f / 0.08f};

// params layout per batch (16 floats):
// [0..2] effective WB gains (gain/norm), [3] 2^exposure, [4] contrast factor,
// [5..8] region strengths (h,s,w,b), [9] vib, [10] sat
__global__ void k_params(const float* __restrict__ tn, const float* __restrict__ ti,
                         const float* __restrict__ ex, const float* __restrict__ co,
                         const float* __restrict__ hi, const float* __restrict__ sh,
                         const float* __restrict__ wh, const float* __restrict__ bl,
                         const float* __restrict__ vi, const float* __restrict__ sa,
                         float* __restrict__ params) {
  const int b = threadIdx.x;
  if (b >= BATCH) return;
  float temp = 2000.0f + 24000.0f * (tn[b] + 1.0f);   // denorm Temperature
  temp = fminf(fmaxf(temp, 2000.0f), 50000.0f);
  const float tr = 6500.0f / temp;
  float red = sqrtf(tr);
  float blue = 1.0f / sqrtf(tr);
  const float ts = fminf(fmaxf(ti[b], -1.5f), 1.5f);  // (150*v)/150 clipped
  const float green = 1.0f - 0.1f * ts;
  red *= (1.0f + 0.05f * ts);
  blue *= (1.0f - 0.05f * ts);
  const float norm = fmaxf(red, fmaxf(green, blue));
  const float inv = 1.0f / fmaxf(norm, 1e-4f);
  float* q = params + b * 16;
  q[0] = red * inv; q[1] = green * inv; q[2] = blue * inv;
  q[3] = exp2f(5.0f * ex[b]);   // 2^exposure, exposure = 5*v
  q[4] = 1.0f + co[b];          // contrast factor, contrast = 100*v
  q[5] = hi[b]; q[6] = sh[b]; q[7] = wh[b]; q[8] = bl[b];
  q[9] = vi[b]; q[10] = sa[b];
}

// interpolate [B,256] control points -> [B,1024] curve (align_corners)
__global__ void k_curve(const float* __restrict__ tc, float* __restrict__ curve) {
  const int b = blockIdx.x;
  const float* src = tc + b * TCL;
  float* dst = curve + b * RES;
  for (int j = threadIdx.x; j < RES; j += BLOCK) {
    const float s = (float)j * ((float)(TCL - 1) / (float)(RES - 1));
    int i0 = (int)s;
    if (i0 > TCL - 1) i0 = TCL - 1;
    int i1 = i0 + 1; if (i1 > TCL - 1) i1 = TCL - 1;
    const float w = s - (float)i0;
    dst[j] = src[i0] * (1.0f - w) + src[i1] * w;
  }
}

// pass A: base pipeline -> luma1 (stored, RT -> stays in L2),
// partial sums of sigmoid mask 0. Image reads tagged non-temporal.
__global__ void k_luma0(const float* __restrict__ img, const float* __restrict__ pp,
                        float* __restrict__ luma, float* __restrict__ part) {
  const int b = blockIdx.y;
  const int q = blockIdx.x * BLOCK + threadIdx.x;
  const float* pr = pp + b * 16;
  const float gr = pr[0], gg = pr[1], gb = pr[2], ex = pr[3], cf = pr[4];
  const size_t base = (size_t)b * 3 * P + (size_t)q * 4;
  const v4f r4 = ntload4(img + base);
  const v4f g4 = ntload4(img + base + P);
  const v4f b4 = ntload4(img + base + 2 * (size_t)P);
  v4f L;
  float s = 0.0f;
  #pragma unroll
  for (int k = 0; k < 4; ++k) {
    const float r = basech(r4[k], gr, ex, cf);
    const float g = basech(g4[k], gg, ex, cf);
    const float bb = basech(b4[k], gb, ex, cf);
    const float l = clip01(lumf(r, g, bb));
    L[k] = l;
    s += fsig((l - 0.7f) * 10.0f);
  }
  *(v4f*)(luma + (size_t)b * P + (size_t)q * 4) = L;  // RT: keep in L2
  block_reduce_store(s, part + (size_t)b * NBLK + blockIdx.x);
}

// deterministic reduction of per-block partials -> mean for `stage`
__global__ void k_reduce(const float* __restrict__ part, float* __restrict__ means,
                         int stage) {
  const int b = blockIdx.x;
  float s = 0.0f;
  for (int i = threadIdx.x; i < NBLK; i += BLOCK) s += part[(size_t)b * NBLK + i];
  __shared__ float sm[BLOCK];
  sm[threadIdx.x] = s;
  __syncthreads();
  #pragma unroll
  for (int off = BLOCK / 2; off > 0; off >>= 1) {
    if (threadIdx.x < off) sm[threadIdx.x] += sm[threadIdx.x + off];
    __syncthreads();
  }
  if (threadIdx.x == 0) means[stage * BATCH + b] = sm[0] * (1.0f / (float)P);
}

// chain luma through regions [0..S), accumulate sigmoid mask S.
// luma loads use default RT policy -> should hit L2 (50 MB resident).
template <int S>
__global__ void k_mask(const float* __restrict__ luma, const float* __restrict__ pp,
                       const float* __restrict__ means, float* __restrict__ part) {
  const int b = blockIdx.y;
  const int q = blockIdx.x * BLOCK + threadIdx.x;
  const float* pr = pp + b * 16;
  float st[4], mn[4];
  #pragma unroll
  for (int j = 0; j < S; ++j) { st[j] = pr[5 + j]; mn[j] = means[j * BATCH + b]; }
  const v4f l4 = *(const v4f*)(luma + (size_t)b * P + (size_t)q * 4);
  float s = 0.0f;
  #pragma unroll
  for (int k = 0; k < 4; ++k) {
    float l = l4[k];
    #pragma unroll
    for (int j = 0; j < S; ++j) {
      const float m = fsig((l - c_piv[j]) * c_iw[j]);
      l = clip01(l + st[j] * (m - mn[j]));
    }
    s += fsig((l - c_piv[S]) * c_iw[S]);
  }
  block_reduce_store(s, part + (size_t)b * NBLK + blockIdx.x);
}

// final fused pass: base -> 4 regions -> tone curve (LDS LUT) -> vibrance -> sat
// Image reads + output writes non-temporal (pure streaming).
__global__ void k_final(const float* __restrict__ img, const float* __restrict__ pp,
                        const float* __restrict__ means, const float* __restrict__ curve,
                        float* __restrict__ out) {
  __shared__ float sc[RES];
  const int b = blockIdx.y;
  const float* gc = curve + (size_t)b * RES;
#ifdef USE_ASYNC_LDS
  {
    // 256 threads x 16B = 4KB: one async b128 per thread
    __builtin_amdgcn_global_load_async_to_lds_b128(
        (async_g_t)(gc + threadIdx.x * 4), (async_l_t)(sc + threadIdx.x * 4), 0, 0);
    __builtin_amdgcn_s_wait_asynccnt(0);
  }
  __syncthreads();
#else
  for (int i = threadIdx.x; i < RES; i += BLOCK) sc[i] = gc[i];
  __syncthreads();
#endif

  const float* pr = pp + b * 16;
  const float gr = pr[0], gg = pr[1], gb = pr[2], ex = pr[3], cf = pr[4];
  const float st[4] = {pr[5], pr[6], pr[7], pr[8]};
  const float vib = pr[9], sat = pr[10];
  const float mn[4] = {means[0 * BATCH + b], means[1 * BATCH + b],
                       means[2 * BATCH + b], means[3 * BATCH + b]};

  const int q = blockIdx.x * BLOCK + threadIdx.x;
  const size_t base = (size_t)b * 3 * P + (size_t)q * 4;
  const v4f r4 = ntload4(img + base);
  const v4f g4 = ntload4(img + base + P);
  const v4f b4 = ntload4(img + base + 2 * (size_t)P);
  v4f Ro, Go, Bo;

  #pragma unroll
  for (int k = 0; k < 4; ++k) {
    float r = basech(r4[k], gr, ex, cf);
    float g = basech(g4[k], gg, ex, cf);
    float bb = basech(b4[k], gb, ex, cf);
    float l = clip01(lumf(r, g, bb));
    // four regions
    #pragma unroll
    for (int j = 0; j < 4; ++j) {
      const float m = fsig((l - c_piv[j]) * c_iw[j]);
      const float nl = clip01(l + st[j] * (m - mn[j]));
      const float ratio = (l > 1e-4f) ? nl * frcp(l) : 1.0f;
      r = clip01(r * ratio);
      g = clip01(g * ratio);
      bb = clip01(bb * ratio);
      l = nl;
    }
    // tone curve (LDS LUT)
    {
      const float tl = clip01(lumf(r, g, bb));
      const float coords = tl * (float)(RES - 1);
      int il = (int)coords;
      if (il > RES - 1) il = RES - 1;
      int ih = il + 1; if (ih > RES - 1) ih = RES - 1;
      const float w = coords - (float)il;
      const float target = sc[il] * (1.0f - w) + sc[ih] * w;
      const float ratio = (tl > 1e-5f) ? target * frcp(tl) : 1.0f;
      r = clip01(r * ratio);
      g = clip01(g * ratio);
      bb = clip01(bb * ratio);
    }
    // vibrance
    {
      const float lv = lumf(r, g, bb);
      const float cr = r - lv, cg = g - lv, cb = bb - lv;
      const float cn = sqrtf(cr * cr + cg * cg + cb * cb + 1e-6f);
      const float msk = __expf(-4.0f * cn);
      const float gn = fminf(fmaxf(1.0f + vib * msk, 0.2f), 4.0f);
      r = clip01(lv + cr * gn);
      g = clip01(lv + cg * gn);
      bb = clip01(lv + cb * gn);
    }
    // saturation
    {
      const float ls = lumf(r, g, bb);
      const float gs = 1.0f + sat;
      r = clip01(ls + (r - ls) * gs);
      g = clip01(ls + (g - ls) * gs);
      bb = clip01(ls + (bb - ls) * gs);
    }
    Ro[k] = r; Go[k] = g; Bo[k] = bb;
  }

  ntstore4(out + base, Ro);
  ntstore4(out + base + P, Go);
  ntstore4(out + base + 2 * (size_t)P, Bo);
}

extern "C" void kernel_launch(void* const* d_in, const int* in_sizes, int n_in,
                              void* d_out, int out_size, void* d_ws, size_t ws_size,
                              hipStream_t stream) {
  const float* img = (const float*)d_in[0];
  const float* tc  = (const float*)d_in[1];
  // workspace layout (floats)
  float* ws = (float*)d_ws;
  float* luma   = ws;                                   // B*P
  float* part   = luma + (size_t)BATCH * P;             // B*NBLK
  float* means  = part + (size_t)BATCH * NBLK;          // 4*B
  float* params = means + 4 * BATCH;                    // 16*B
  float* curve  = params + 16 * BATCH;                  // B*RES

  k_params<<<1, 32, 0, stream>>>((const float*)d_in[2], (const float*)d_in[3],
                                 (const float*)d_in[4], (const float*)d_in[5],
                                 (const float*)d_in[6], (const float*)d_in[7],
                                 (const float*)d_in[8], (const float*)d_in[9],
                                 (const float*)d_in[10], (const float*)d_in[11],
                                 params);
  k_curve<<<BATCH, BLOCK, 0, stream>>>(tc, curve);

  const dim3 grid(NBLK, BATCH);
  k_luma0<<<grid, BLOCK, 0, stream>>>(img, params, luma, part);
  k_reduce<<<BATCH, BLOCK, 0, stream>>>(part, means, 0);
  k_mask<1><<<grid, BLOCK, 0, stream>>>(luma, params, means, part);
  k_reduce<<<BATCH, BLOCK, 0, stream>>>(part, means, 1);
  k_mask<2><<<grid, BLOCK, 0, stream>>>(luma, params, means, part);
  k_reduce<<<BATCH, BLOCK, 0, stream>>>(part, means, 2);
  k_mask<3><<<grid, BLOCK, 0, stream>>>(luma, params, means, part);
  k_reduce<<<BATCH, BLOCK, 0, stream>>>(part, means, 3);

  k_final<<<grid, BLOCK, 0, stream>>>(img, params, means, curve, (float*)d_out);
}